// JavaEncoder_10075993276850
// MI455X (gfx1250) — compile-verified
//
#include <hip/hip_runtime.h>
#include <math.h>

#define NN   200000
#define NE   1600000
#define NFEAT 128
#define DIM  64
#define NB   256

#define GRU_LDS_STRIDE 68          // 64 + 4 pad -> conflict-free ds_load_b64
#define GRU_LDS_ROWS   (6 * DIM)   // 192 rows wih + 192 rows whh

typedef __attribute__((ext_vector_type(2))) float v2f;
typedef __attribute__((ext_vector_type(8))) float v8f;

__device__ __forceinline__ v8f wmma4(v2f a, v2f b, v8f c) {
  // V_WMMA_F32_16X16X4_F32: D = A(16x4,f32) * B(4x16,f32) + C(16x16,f32)
  return __builtin_amdgcn_wmma_f32_16x16x4_f32(false, a, false, b, (short)0, c,
                                               false, false);
}

__device__ __forceinline__ float sigf(float x) { return 1.0f / (1.0f + expf(-x)); }

__device__ __forceinline__ void atomicMaxF(float* addr, float val) {
  int* ia = (int*)addr;
  int old = __float_as_int(*addr);
  while (__int_as_float(old) < val) {
    int assumed = old;
    old = atomicCAS(ia, assumed, __float_as_int(val));
    if (old == assumed) break;
  }
}

// ---------------- utility ----------------
__global__ void k_zero(float* p, int n) {
  int i = blockIdx.x * blockDim.x + threadIdx.x;
  if (i < n) p[i] = 0.0f;
}

// ---------------- degree / norm ----------------
__global__ void k_deg(const int* __restrict__ col, float* __restrict__ deg) {
  int i = blockIdx.x * blockDim.x + threadIdx.x;
  if (i < NE) atomicAdd(&deg[col[i]], 1.0f);
}

__global__ void k_dinv(float* __restrict__ d) {
  int i = blockIdx.x * blockDim.x + threadIdx.x;
  if (i < NN) {
    float v = d[i] + 1.0f;  // + self loop
    d[i] = rsqrtf(v);       // deg >= 1 always
  }
}

// ---------------- generic GEMM: C[M x 64] = act(A[M x K] @ B[K x 64] + bias) ----
// block = 256 threads = 8 waves; wave => one 16x16 tile; block => 32 rows x 64 cols
__global__ void k_gemm64(const float* __restrict__ A, int K,
                         const float* __restrict__ Bm,
                         const float* __restrict__ bias,
                         float* __restrict__ C, int relu) {
  int lane = threadIdx.x & 31;
  int wv   = threadIdx.x >> 5;
  int rt   = blockIdx.x * 2 + (wv >> 2);
  int ct   = wv & 3;
  int nl   = lane & 15;
  int hi   = lane >> 4;
  int arow = rt * 16 + nl;
  int bcol = ct * 16 + nl;
  v8f acc = {0.f, 0.f, 0.f, 0.f, 0.f, 0.f, 0.f, 0.f};
  for (int kk = 0; kk < K; kk += 4) {
    int k0 = kk + 2 * hi;
    v2f av, bv;
    av.x = A[arow * K + k0];
    av.y = A[arow * K + k0 + 1];
    bv.x = Bm[(k0) * DIM + bcol];
    bv.y = Bm[(k0 + 1) * DIM + bcol];
    acc = wmma4(av, bv, acc);
  }
  float bb = bias ? bias[bcol] : 0.0f;
#pragma unroll
  for (int v = 0; v < 8; v++) {
    int row = rt * 16 + v + 8 * hi;
    float val = acc[v] + bb;
    if (relu) val = fmaxf(val, 0.0f);
    C[row * DIM + bcol] = val;
  }
}

// ---------------- message init: self-loop + bias ----------------
__global__ void k_minit(const float* __restrict__ hw, const float* __restrict__ dinv,
                        const float* __restrict__ convb, float* __restrict__ m) {
  int t = blockIdx.x * blockDim.x + threadIdx.x;  // < NN*DIM
  int n = t >> 6, d = t & 63;
  float di = dinv[n];
  m[t] = di * di * hw[t] + convb[d];
}

// ---------------- edge scatter: m[col] += dinv[row]*dinv[col]*hw[row] ---------
__global__ void k_edge(const int* __restrict__ rowi, const int* __restrict__ coli,
                       const float* __restrict__ dinv, const float* __restrict__ hw,
                       float* __restrict__ m) {
  int t = blockIdx.x * blockDim.x + threadIdx.x;  // < NE*16
  int e = t >> 4, part = t & 15;
  int r = rowi[e], c = coli[e];
  float nrm = dinv[r] * dinv[c];
  const float4 v = *(const float4*)(hw + r * DIM + part * 4);
  float* dst = m + c * DIM + part * 4;
  atomicAdd(dst + 0, nrm * v.x);
  atomicAdd(dst + 1, nrm * v.y);
  atomicAdd(dst + 2, nrm * v.z);
  atomicAdd(dst + 3, nrm * v.w);
}

// ---------------- fused GRU cell (relu(m) as input), LDS-staged weights -------
// block = 512 threads = 16 waves: 4 node-tiles x 4 col-chunks; grid = NN/64
// dynamic LDS: 384 rows x 68 floats (rows 0..191 = wih, 192..383 = whh)
__global__ void k_gru(const float* __restrict__ m, float* __restrict__ h,
                      const float* __restrict__ wih, const float* __restrict__ whh,
                      const float* __restrict__ bih, const float* __restrict__ bhh) {
  extern __shared__ float s_w[];
  for (int idx = threadIdx.x; idx < 2 * 3 * DIM * DIM; idx += blockDim.x) {
    int r = idx >> 6, c = idx & 63;  // coalesced global read, padded LDS store
    float v = (r < 3 * DIM) ? wih[r * DIM + c] : whh[(r - 3 * DIM) * DIM + c];
    s_w[r * GRU_LDS_STRIDE + c] = v;
  }
  __syncthreads();

  int lane = threadIdx.x & 31;
  int w    = threadIdx.x >> 5;  // 0..15
  int tl   = w >> 2;            // node tile within block
  int wv   = w & 3;             // col chunk
  int nl   = lane & 15;
  int hi   = lane >> 4;
  int tile = blockIdx.x * 4 + tl;
  int arow = tile * 16 + nl;
  int outc = wv * 16 + nl;
  const float* sih = s_w;
  const float* shh = s_w + 3 * DIM * GRU_LDS_STRIDE;

  v8f ar = {0.f,0.f,0.f,0.f,0.f,0.f,0.f,0.f};
  v8f az = ar, ai = ar, an = ar;
  for (int kk = 0; kk < DIM; kk += 4) {
    int k0 = kk + 2 * hi;
    v2f am, ah, b;
    am.x = fmaxf(m[arow * DIM + k0], 0.0f);
    am.y = fmaxf(m[arow * DIM + k0 + 1], 0.0f);
    ah.x = h[arow * DIM + k0];
    ah.y = h[arow * DIM + k0 + 1];
    // reset gate: m@Wir^T + h@Whr^T  (same accumulator)
    b.x = sih[outc * GRU_LDS_STRIDE + k0];
    b.y = sih[outc * GRU_LDS_STRIDE + k0 + 1];
    ar = wmma4(am, b, ar);
    b.x = shh[outc * GRU_LDS_STRIDE + k0];
    b.y = shh[outc * GRU_LDS_STRIDE + k0 + 1];
    ar = wmma4(ah, b, ar);
    // update gate
    b.x = sih[(DIM + outc) * GRU_LDS_STRIDE + k0];
    b.y = sih[(DIM + outc) * GRU_LDS_STRIDE + k0 + 1];
    az = wmma4(am, b, az);
    b.x = shh[(DIM + outc) * GRU_LDS_STRIDE + k0];
    b.y = shh[(DIM + outc) * GRU_LDS_STRIDE + k0 + 1];
    az = wmma4(ah, b, az);
    // candidate: input part and hidden part kept separate
    b.x = sih[(2 * DIM + outc) * GRU_LDS_STRIDE + k0];
    b.y = sih[(2 * DIM + outc) * GRU_LDS_STRIDE + k0 + 1];
    ai = wmma4(am, b, ai);
    b.x = shh[(2 * DIM + outc) * GRU_LDS_STRIDE + k0];
    b.y = shh[(2 * DIM + outc) * GRU_LDS_STRIDE + k0 + 1];
    an = wmma4(ah, b, an);
  }

  float hold[8], hnew[8];
#pragma unroll
  for (int v = 0; v < 8; v++) {
    int row = tile * 16 + v + 8 * hi;
    hold[v] = h[row * DIM + outc];
  }
  float brz0 = bih[outc] + bhh[outc];
  float brz1 = bih[DIM + outc] + bhh[DIM + outc];
  float bi2  = bih[2 * DIM + outc];
  float bh2  = bhh[2 * DIM + outc];
#pragma unroll
  for (int v = 0; v < 8; v++) {
    float r  = sigf(ar[v] + brz0);
    float z  = sigf(az[v] + brz1);
    float nc = tanhf(ai[v] + bi2 + r * (an[v] + bh2));
    hnew[v] = (1.0f - z) * nc + z * hold[v];
  }
  __syncthreads();  // all reads of h done before any wave writes its col chunk
#pragma unroll
  for (int v = 0; v < 8; v++) {
    int row = tile * 16 + v + 8 * hi;
    h[row * DIM + outc] = hnew[v];
  }
}

// ---------------- LSTM gates: gates[256,256] = qstar@Wih^T + hl@Whh^T + b ----
// grid = 32 blocks x 8 waves = 256 tiles of 16x16
__global__ void k_lstm_gates(const float* __restrict__ qstar, const float* __restrict__ hl,
                             const float* __restrict__ wih, const float* __restrict__ whh,
                             const float* __restrict__ bih, const float* __restrict__ bhh,
                             float* __restrict__ gates) {
  int lane = threadIdx.x & 31;
  int wv   = threadIdx.x >> 5;
  int tid  = blockIdx.x * 8 + wv;  // 0..255
  int rt   = tid >> 4;
  int ct   = tid & 15;
  int nl   = lane & 15;
  int hi   = lane >> 4;
  int arow = rt * 16 + nl;
  int gcol = ct * 16 + nl;  // 0..255
  v8f acc = {0.f,0.f,0.f,0.f,0.f,0.f,0.f,0.f};
  for (int kk = 0; kk < 2 * DIM; kk += 4) {
    int k0 = kk + 2 * hi;
    v2f a, b;
    a.x = qstar[arow * (2 * DIM) + k0];
    a.y = qstar[arow * (2 * DIM) + k0 + 1];
    b.x = wih[gcol * (2 * DIM) + k0];
    b.y = wih[gcol * (2 * DIM) + k0 + 1];
    acc = wmma4(a, b, acc);
  }
  for (int kk = 0; kk < DIM; kk += 4) {
    int k0 = kk + 2 * hi;
    v2f a, b;
    a.x = hl[arow * DIM + k0];
    a.y = hl[arow * DIM + k0 + 1];
    b.x = whh[gcol * DIM + k0];
    b.y = whh[gcol * DIM + k0 + 1];
    acc = wmma4(a, b, acc);
  }
  float bb = bih[gcol] + bhh[gcol];
#pragma unroll
  for (int v = 0; v < 8; v++) {
    int row = rt * 16 + v + 8 * hi;
    gates[row * (4 * DIM) + gcol] = acc[v] + bb;
  }
}

__global__ void k_lstm_ew(const float* __restrict__ gates, float* __restrict__ hl,
                          float* __restrict__ cl, float* __restrict__ qstar) {
  int t = blockIdx.x * blockDim.x + threadIdx.x;  // < NB*DIM
  int b = t >> 6, d = t & 63;
  float ig = gates[b * 4 * DIM + d];
  float fg = gates[b * 4 * DIM + DIM + d];
  float gg = gates[b * 4 * DIM + 2 * DIM + d];
  float og = gates[b * 4 * DIM + 3 * DIM + d];
  float c  = sigf(fg) * cl[t] + sigf(ig) * tanhf(gg);
  float hh = sigf(og) * tanhf(c);
  cl[t] = c;
  hl[t] = hh;
  qstar[b * 2 * DIM + d] = hh;  // q part of q_star
}

// ---------------- attention ----------------
__global__ void k_attn_init(float* __restrict__ emax, float* __restrict__ denom,
                            float* __restrict__ qstar) {
  int t = blockIdx.x * blockDim.x + threadIdx.x;  // < NB*DIM
  int b = t >> 6, d = t & 63;
  qstar[b * 2 * DIM + DIM + d] = 0.0f;  // r_read accumulator
  if (d == 0) {
    emax[b]  = -__builtin_inff();
    denom[b] = 0.0f;
  }
}

__global__ void k_attn_e(const float* __restrict__ out, const float* __restrict__ hl,
                         const int* __restrict__ batch, float* __restrict__ e,
                         float* __restrict__ emax) {
  int n = blockIdx.x * blockDim.x + threadIdx.x;
  if (n >= NN) return;
  int b = batch[n];
  float s = 0.0f;
#pragma unroll 8
  for (int d = 0; d < DIM; d++) s += out[n * DIM + d] * hl[b * DIM + d];
  e[n] = s;
  atomicMaxF(&emax[b], s);
}

__global__ void k_attn_a(const float* __restrict__ e, const int* __restrict__ batch,
                         const float* __restrict__ emax, float* __restrict__ a,
                         float* __restrict__ denom) {
  int n = blockIdx.x * blockDim.x + threadIdx.x;
  if (n >= NN) return;
  int b = batch[n];
  float v = expf(e[n] - emax[b]);
  a[n] = v;
  atomicAdd(&denom[b], v);
}

// block = 256 threads = 16 nodes x 16 parts; batch is sorted, so blocks are
// almost always batch-uniform -> reduce in LDS, 64 global atomics per block.
__global__ void k_attn_r(const float* __restrict__ out, const float* __restrict__ a,
                         const float* __restrict__ denom, const int* __restrict__ batch,
                         float* __restrict__ qstar) {
  __shared__ float sacc[DIM];
  int t = blockIdx.x * blockDim.x + threadIdx.x;  // < NN*16
  int n = t >> 4, part = t & 15;
  int nodeBase = blockIdx.x * 16;
  int b0  = batch[nodeBase];
  int b15 = batch[nodeBase + 15];
  if (threadIdx.x < DIM) sacc[threadIdx.x] = 0.0f;
  __syncthreads();
  int b = batch[n];
  float w = a[n] / denom[b];
  const float4 v = *(const float4*)(out + n * DIM + part * 4);
  if (b0 == b15) {  // block-uniform branch
    atomicAdd(&sacc[part * 4 + 0], w * v.x);
    atomicAdd(&sacc[part * 4 + 1], w * v.y);
    atomicAdd(&sacc[part * 4 + 2], w * v.z);
    atomicAdd(&sacc[part * 4 + 3], w * v.w);
    __syncthreads();
    if (threadIdx.x < DIM)
      atomicAdd(&qstar[b0 * 2 * DIM + DIM + threadIdx.x], sacc[threadIdx.x]);
  } else {
    float* dst = qstar + b * 2 * DIM + DIM + part * 4;
    atomicAdd(dst + 0, w * v.x);
    atomicAdd(dst + 1, w * v.y);
    atomicAdd(dst + 2, w * v.z);
    atomicAdd(dst + 3, w * v.w);
  }
}

// ---------------- driver ----------------
extern "C" void kernel_launch(void* const* d_in, const int* in_sizes, int n_in,
                              void* d_out, int out_size, void* d_ws, size_t ws_size,
                              hipStream_t stream) {
  const float* x         = (const float*)d_in[0];
  const int*   ei        = (const int*)d_in[1];
  const int*   row       = ei;
  const int*   col       = ei + NE;
  const int*   batch     = (const int*)d_in[2];
  const float* lin0_w    = (const float*)d_in[3];
  const float* lin0_b    = (const float*)d_in[4];
  const float* conv_w    = (const float*)d_in[5];
  const float* conv_b    = (const float*)d_in[6];
  const float* gru_w_ih  = (const float*)d_in[7];
  const float* gru_w_hh  = (const float*)d_in[8];
  const float* gru_b_ih  = (const float*)d_in[9];
  const float* gru_b_hh  = (const float*)d_in[10];
  const float* lstm_w_ih = (const float*)d_in[11];
  const float* lstm_w_hh = (const float*)d_in[12];
  const float* lstm_b_ih = (const float*)d_in[13];
  const float* lstm_b_hh = (const float*)d_in[14];

  float* ws    = (float*)d_ws;
  float* dinv  = ws;                    // N
  float* hw    = dinv + NN;             // N*64
  float* m     = hw + NN * DIM;         // N*64
  float* ev    = m + NN * DIM;          // N
  float* av    = ev + NN;               // N
  float* gates = av + NN;               // B*256
  float* hl    = gates + NB * 4 * DIM;  // B*64
  float* cl    = hl + NB * DIM;         // B*64
  float* emax  = cl + NB * DIM;         // B
  float* denom = emax + NB;             // B

  float* qstar = (float*)d_out;            // [B, 2*DIM]
  float* outf  = qstar + NB * 2 * DIM;     // feat buffer [N, DIM] (== out/h)

  // --- gcn_norm: deg = in-degree + 1 (self loop), dinv = rsqrt(deg) ---
  k_zero<<<(NN + 255) / 256, 256, 0, stream>>>(dinv, NN);
  k_deg<<<(NE + 255) / 256, 256, 0, stream>>>(col, dinv);
  k_dinv<<<(NN + 255) / 256, 256, 0, stream>>>(dinv);

  // --- out = relu(x @ lin0_w + lin0_b) ---
  k_gemm64<<<NN / 32, 256, 0, stream>>>(x, NFEAT, lin0_w, lin0_b, outf, 1);

  // --- 3x (GCNConv + GRU) ---
  size_t gruLds = (size_t)GRU_LDS_ROWS * GRU_LDS_STRIDE * sizeof(float);  // ~102KB
  for (int it = 0; it < 3; ++it) {
    k_gemm64<<<NN / 32, 256, 0, stream>>>(outf, DIM, conv_w, nullptr, hw, 0);
    k_minit<<<(NN * DIM) / 256, 256, 0, stream>>>(hw, dinv, conv_b, m);
    k_edge<<<(NE * 16) / 256, 256, 0, stream>>>(row, col, dinv, hw, m);
    k_gru<<<NN / 64, 512, gruLds, stream>>>(m, outf, gru_w_ih, gru_w_hh,
                                            gru_b_ih, gru_b_hh);
  }

  // --- Set2Set init ---
  k_zero<<<(NB * 2 * DIM + 255) / 256, 256, 0, stream>>>(qstar, NB * 2 * DIM);
  k_zero<<<(NB * DIM + 255) / 256, 256, 0, stream>>>(hl, NB * DIM);
  k_zero<<<(NB * DIM + 255) / 256, 256, 0, stream>>>(cl, NB * DIM);

  // --- 3 processing steps ---
  for (int st = 0; st < 3; ++st) {
    k_lstm_gates<<<32, 256, 0, stream>>>(qstar, hl, lstm_w_ih, lstm_w_hh,
                                         lstm_b_ih, lstm_b_hh, gates);
    k_lstm_ew<<<(NB * DIM) / 256, 256, 0, stream>>>(gates, hl, cl, qstar);
    k_attn_init<<<(NB * DIM) / 256, 256, 0, stream>>>(emax, denom, qstar);
    k_attn_e<<<(NN + 255) / 256, 256, 0, stream>>>(outf, hl, batch, ev, emax);
    k_attn_a<<<(NN + 255) / 256, 256, 0, stream>>>(ev, batch, emax, av, denom);
    k_attn_r<<<(NN * 16) / 256, 256, 0, stream>>>(outf, av, denom, batch, qstar);
  }
}